// three_dim_Layer_24747601560316
// MI455X (gfx1250) — compile-verified
//
#include <hip/hip_runtime.h>
#include <hip/hip_bf16.h>
#include <math.h>

#define C_CH 3
#define H_DIM 32
#define W_DIM 32
#define IN_FEATURE (C_CH * H_DIM * W_DIM)   // 3072
#define PAD_W 34
#define PAD_PLANE (PAD_W * PAD_W)           // 1156
#define PAD_TOTAL (C_CH * PAD_PLANE)        // 3468
#define CLAMP_V 0.7f

typedef float v2f __attribute__((ext_vector_type(2)));
typedef float v8f __attribute__((ext_vector_type(8)));

__global__ __launch_bounds__(256)
void snn_fused_kernel(const float* __restrict__ x,
                      const float* __restrict__ tau_m1,
                      const float* __restrict__ tau_s1,
                      const float* __restrict__ tau_sm1,
                      const float* __restrict__ gk,     // (3,3,3,3) OIHW flat = 81
                      const float* __restrict__ gbias,  // (3,)
                      const int*   __restrict__ wexp,   // scalar
                      float* __restrict__ out)
{
    __shared__ float yLds[PAD_TOTAL]; // padded (3,34,34) image, halo stays 0
    __shared__ float wLds[81];
    __shared__ float bLds[3];

    const int tid  = threadIdx.x;
    const int b    = blockIdx.x;
    const int lane = tid & 31;
    const int wv   = tid >> 5;

    // ---- zero padded LDS image + stage conv weights/bias ----
    for (int i = tid; i < PAD_TOTAL; i += 256) yLds[i] = 0.0f;
    if (tid < 81) wLds[tid] = gk[tid];
    if (tid < 3)  bLds[tid] = gbias[tid];
    __syncthreads();

    // ---- phase 1: pointwise transform, scatter into padded LDS ----
    const float spikeScale = exp2f(-(float)(*wexp));
    const float4* x4 = reinterpret_cast<const float4*>(x) + (size_t)b * IN_FEATURE;
    #pragma unroll
    for (int it = 0; it < IN_FEATURE / 256; ++it) {
        const int f = tid + it * 256;
        float4 v = x4[f];
        float elu = (v.x > 0.0f) ? v.x : expm1f(v.x);
        float lrl = (v.y >= 0.0f) ? v.y : 0.01f * v.y; // placeholder (fixed below)
        // note: x order is (x1,x2,x3,reset) = (v.x,v.y,v.z,v.w)
        lrl = (v.z >= 0.0f) ? v.z : 0.01f * v.z;
        float y = elu * tau_m1[f] - v.y * tau_s1[f] + lrl * tau_sm1[f];
        if (v.w > 0.0f) y -= spikeScale;
        const int ci  = f >> 10;
        const int pix = f & 1023;
        const int h   = pix >> 5;
        const int w   = pix & 31;
        yLds[ci * PAD_PLANE + (h + 1) * PAD_W + (w + 1)] = y;
    }
    __syncthreads();

    // ---- phase 2: implicit-GEMM 3x3 conv via V_WMMA_F32_16X16X4_F32 ----
    // N = lane&15 (out channel, 3 real), lane>>4 selects K-pair {0,1} vs {2,3}
    const int n     = lane & 15;
    const int kOff  = (lane >> 4) << 1;   // 0 or 2
    const bool nOK  = (n < 3);

    // B fragments: 7 chunks of K=4, B[k',n] = gk[n*27 + k'] (0 padded)
    v2f bfrag[7];
    #pragma unroll
    for (int cch = 0; cch < 7; ++cch) {
        int k0 = cch * 4 + kOff;
        float b0 = (nOK && k0     < 27) ? wLds[n * 27 + k0]     : 0.0f;
        float b1 = (nOK && k0 + 1 < 27) ? wLds[n * 27 + k0 + 1] : 0.0f;
        bfrag[cch].x = b0;
        bfrag[cch].y = b1;
    }

    const int m    = lane & 15;           // pixel within group (A-matrix row)
    const int mOff = (lane >> 4) << 3;    // C rows 0..7 vs 8..15
    float* outB = out + (size_t)b * IN_FEATURE;

    // 64 groups of 16 contiguous pixels (half-rows); 8 waves x 8 groups
    for (int g = wv; g < 64; g += 8) {
        const int pixBase = g << 4;
        const int p  = pixBase + m;
        const int h  = p >> 5;            // constant across the group
        const int wc = p & 31;

        v8f acc = {};
        #pragma unroll
        for (int cch = 0; cch < 7; ++cch) {
            v2f a;
            #pragma unroll
            for (int j = 0; j < 2; ++j) {
                const int kk = cch * 4 + kOff + j;
                float val = 0.0f;
                if (kk < 27) {
                    const int ci = kk / 9;
                    const int r  = kk - ci * 9;
                    const int dh = r / 3;         // 0..2  (h+dh-1 folded into +1 pad)
                    const int dw = r - dh * 3;    // 0..2
                    val = yLds[ci * PAD_PLANE + (h + dh) * PAD_W + (wc + dw)];
                }
                if (j == 0) a.x = val; else a.y = val;
            }
            acc = __builtin_amdgcn_wmma_f32_16x16x4_f32(
                false, a, false, bfrag[cch], (short)0, acc, false, false);
        }

        // C layout: lane holds column n, rows mOff..mOff+7 in acc[0..7]
        if (nOK) {
            const float bn = bLds[n];
            #pragma unroll
            for (int v = 0; v < 8; ++v) {
                float val = acc[v] + bn;
                val = fminf(CLAMP_V, fmaxf(-CLAMP_V, val));
                outB[n * 1024 + pixBase + mOff + v] = val;
            }
        }
    }
}

extern "C" void kernel_launch(void* const* d_in, const int* in_sizes, int n_in,
                              void* d_out, int out_size, void* d_ws, size_t ws_size,
                              hipStream_t stream) {
    const float* x       = (const float*)d_in[0];
    const float* tau_m1  = (const float*)d_in[1];
    const float* tau_s1  = (const float*)d_in[2];
    const float* tau_sm1 = (const float*)d_in[3];
    const float* gk      = (const float*)d_in[4];
    const float* gbias   = (const float*)d_in[5];
    const int*   wexp    = (const int*)d_in[6];
    float* out = (float*)d_out;

    const int nB = in_sizes[0] / (IN_FEATURE * 4);
    snn_fused_kernel<<<dim3(nB), dim3(256), 0, stream>>>(
        x, tau_m1, tau_s1, tau_sm1, gk, gbias, wexp, out);
}